// Mask3DLoss_49959059587445
// MI455X (gfx1250) — compile-verified
//
#include <hip/hip_runtime.h>
#include <hip/hip_bf16.h>
#include <math.h>

typedef __attribute__((ext_vector_type(2))) float v2f;
typedef __attribute__((ext_vector_type(8))) float v8f;

namespace {
constexpr int kB      = 64;
constexpr int kNPred  = 16384;
constexpr int kNGt    = 8192;
constexpr int kNBox   = kB * kNGt;                 // 524288 gt boxes
constexpr int kBoxesPerTile = 5;                   // 3 rows per box in a 16x16 tile
constexpr int kTilesPerIter = 6;
constexpr int kBoxesPerIter = kBoxesPerTile * kTilesPerIter;   // 30
constexpr int kNIters = (kNBox + kBoxesPerIter - 1) / kBoxesPerIter;  // 17477
constexpr int kBlock  = 32;                        // single wave32 workgroup
constexpr int kGridMain  = 4096;                   // 4096 waves
constexpr int kGridExtra = 256;
constexpr int kBlockExtra = 256;
constexpr int kNPart  = kGridMain + kGridExtra;    // 4352 partials

constexpr float kInvReg = 1.0f / (float)(kB * kNGt * 7);   // smooth-L1 mean denom
constexpr float kInvBce = 1.0f / (float)(kB * kNGt);       // BCE mean denom
constexpr float kExtraScale = 0.3f / (float)kB;            // 0.3 * sum / B
}

// ---------------------------------------------------------------------------
// Kernel 1: per-gt-box pipeline, one wave per workgroup.
//   Super-iteration = 30 boxes:
//     phase 1: 180 x float4 coalesced corner loads -> LDS (global_load_b128)
//     phase 2: 90 (box,dim) centers
//     phase 3: 6 x (two V_WMMA_F32_16X16X4_F32, 5 boxes per 16x16 tile,
//              3x3 diagonal blocks = covariances; A==B since cov = d^T d)
//     phase 4: D tiles -> LDS
//     phase 5: 30 lanes eigensolve one box each + smooth-L1/BCE accumulate
//   Single-wave workgroup => __syncthreads() is an S_NOP-class barrier but
//   keeps compiler-level LDS ordering; all comms are intra-wave.
// ---------------------------------------------------------------------------
__global__ __launch_bounds__(kBlock)
void mask3d_gt_kernel(const float* __restrict__ pred_boxes,
                      const float* __restrict__ pred_scores,
                      const float* __restrict__ gt_corners,
                      float* __restrict__ partials) {
  __shared__ float4 s_corn4[192];               // 768 floats (720 used/iter)
  __shared__ float  s_cen[92];                  // 30 boxes * 3 dims
  __shared__ float  s_cov[kTilesPerIter * 256]; // 6 WMMA D tiles
  float* s_corn = (float*)s_corn4;

  const int lane = threadIdx.x;                 // 0..31
  const float4* g4 = (const float4*)gt_corners;
  const int total4 = (kNBox * 24) / 4;          // total float4s in gt_corners

  // Precomputed fragment-lane mapping (constant across iterations).
  const int  mm  = lane & 15;
  const int  kb  = (lane >= 16) ? 2 : 0;        // K pair base for this half-wave
  const bool rok = (mm < 15);                   // row 15 of tile is zero padding
  const int  bq  = rok ? (mm / 3) : 0;          // box-in-tile
  const int  ii  = rok ? (mm - bq * 3) : 0;     // dim 0..2
  const float msk = rok ? 1.0f : 0.0f;
  const int  mrow = (lane >= 16) ? 8 : 0;

  float acc = 0.0f;

  for (int it = blockIdx.x; it < kNIters; it += kGridMain) {
    const int box0 = it * kBoxesPerIter;

    // ---- Phase 1: corners -> LDS, 16B-aligned float4 loads ----
    const int f4base = it * 180;                // 720 floats / 4
#pragma unroll
    for (int e = lane, u = 0; u < 6; ++u, e += 32) {   // 192 loads, uniform
      int idx4 = f4base + e;
      if (idx4 > total4 - 1) idx4 = total4 - 1;
      s_corn4[e] = g4[idx4];
    }
    // CDNA5 prefetch of the next super-iteration's corners
    {
      int nit = it + kGridMain;
      if (nit < kNIters) {
        __builtin_prefetch(gt_corners + (size_t)nit * kBoxesPerIter * 24 + lane * 16, 0, 0);
      }
    }
    __syncthreads();

    // ---- Phase 2: centers (90 box,dim pairs) ----
    for (int pp = lane; pp < 90; pp += 32) {
      int bx = pp / 3, i = pp - bx * 3;
      float s = 0.0f;
#pragma unroll
      for (int k = 0; k < 8; ++k) s += s_corn[bx * 24 + k * 3 + i];
      s_cen[pp] = s * 0.125f;
    }
    __syncthreads();

    // ---- Phase 3/4: 6 tiles x (2 WMMA + spill D) -- EXEC all ones here ----
#pragma unroll
    for (int tt = 0; tt < kTilesPerIter; ++tt) {
      const int bl = tt * kBoxesPerTile + bq;
      const float ce = s_cen[bl * 3 + ii];
      const float* cp = &s_corn[bl * 24 + ii];
      // A 16x4 f32 layout: lanes 0-15 = M rows, v0/v1 = K(kb)/K(kb+1);
      // lanes 16-31 same rows, K(kb=2)/K(3). B mirrors A for a Gram matrix.
      v2f a0, a1;
      a0.x = (cp[(kb + 0) * 3] - ce) * msk;
      a0.y = (cp[(kb + 1) * 3] - ce) * msk;
      a1.x = (cp[(kb + 4) * 3] - ce) * msk;
      a1.y = (cp[(kb + 5) * 3] - ce) * msk;

      v8f c = {0, 0, 0, 0, 0, 0, 0, 0};
      c = __builtin_amdgcn_wmma_f32_16x16x4_f32(false, a0, false, a0, (short)0, c, false, false);
      c = __builtin_amdgcn_wmma_f32_16x16x4_f32(false, a1, false, a1, (short)0, c, false, false);

#pragma unroll
      for (int v = 0; v < 8; ++v) s_cov[tt * 256 + (mrow + v) * 16 + mm] = c[v];
    }
    __syncthreads();

    // ---- Phase 5: 30 lanes, one box each ----
    if (lane < kBoxesPerIter) {
      int gidx = box0 + lane;
      const bool valid = (gidx < kNBox);
      if (gidx >= kNBox) gidx = kNBox - 1;

      const int tt = lane / kBoxesPerTile;
      const int r0 = (lane - tt * kBoxesPerTile) * 3;
      const float* C = &s_cov[tt * 256];
      float a00 = C[(r0 + 0) * 16 + r0 + 0];
      float a01 = C[(r0 + 0) * 16 + r0 + 1];
      float a02 = C[(r0 + 0) * 16 + r0 + 2];
      float a11 = C[(r0 + 1) * 16 + r0 + 1];
      float a12 = C[(r0 + 1) * 16 + r0 + 2];
      float a22 = C[(r0 + 2) * 16 + r0 + 2];
      float cx = s_cen[lane * 3 + 0];
      float cy = s_cen[lane * 3 + 1];
      float cz = s_cen[lane * 3 + 2];

      // Closed-form symmetric 3x3 eigenvalues (trigonometric method)
      float q  = (a00 + a11 + a22) * (1.0f / 3.0f);
      float p1 = a01 * a01 + a02 * a02 + a12 * a12;
      float b00 = a00 - q, b11 = a11 - q, b22 = a22 - q;
      float p2 = b00 * b00 + b11 * b11 + b22 * b22 + 2.0f * p1;
      float p  = sqrtf(p2 * (1.0f / 6.0f)) + 1e-30f;
      float ip = 1.0f / p;
      float d00 = b00 * ip, d01 = a01 * ip, d02 = a02 * ip;
      float d11 = b11 * ip, d12 = a12 * ip, d22 = b22 * ip;
      float det = d00 * (d11 * d22 - d12 * d12)
                - d01 * (d01 * d22 - d12 * d02)
                + d02 * (d01 * d12 - d11 * d02);
      float rr = fminf(1.0f, fmaxf(-1.0f, 0.5f * det));
      float phi = acosf(rr) * (1.0f / 3.0f);
      float l2 = q + 2.0f * p * __cosf(phi);                       // largest
      float l0 = q + 2.0f * p * __cosf(phi + 2.0943951023931953f); // smallest
      float l1 = 3.0f * q - l0 - l2;                               // middle
      // size = 2*sqrt(eigvals)[perm 1,2,0]
      float s0 = 2.0f * sqrtf(fmaxf(l1, 0.0f));
      float s1 = 2.0f * sqrtf(fmaxf(l2, 0.0f));
      float s2 = 2.0f * sqrtf(fmaxf(l0, 0.0f));

      // Eigenvector of smallest eigenvalue: largest cross product of rows
      float m00 = a00 - l0, m11 = a11 - l0, m22 = a22 - l0;
      float c01x = a01 * a12 - a02 * m11, c01y = a02 * a01 - m00 * a12, c01z = m00 * m11 - a01 * a01;
      float c12x = m11 * m22 - a12 * a12, c12y = a12 * a02 - a01 * m22, c12z = a01 * a12 - m11 * a02;
      float c02x = a01 * m22 - a02 * a12, c02y = a02 * a02 - m00 * m22, c02z = m00 * a12 - a01 * a02;
      float n01 = c01x * c01x + c01y * c01y + c01z * c01z;
      float n12 = c12x * c12x + c12y * c12y + c12z * c12z;
      float n02 = c02x * c02x + c02y * c02y + c02z * c02z;
      float vx = c01x, vy = c01y, nb = n01;
      if (n12 > nb) { nb = n12; vx = c12x; vy = c12y; }
      if (n02 > nb) { nb = n02; vx = c02x; vy = c02y; }
      float yaw = atan2f(vy, vx);

      // smooth-L1 vs pred_boxes + BCE on matched score
      const int bidx = gidx / kNGt;
      const int midx = gidx - bidx * kNGt;
      const float* pb = pred_boxes + ((size_t)bidx * kNPred + midx) * 7;
      float prm[7] = {cx, cy, cz, s0, s1, s2, yaw};
      float sl = 0.0f;
#pragma unroll
      for (int j = 0; j < 7; ++j) {
        float dd = fabsf(pb[j] - prm[j]);
        sl += (dd < 1.0f) ? (0.5f * dd * dd) : (dd - 0.5f);
      }
      float sc = pred_scores[(size_t)bidx * kNPred + midx];
      sc = fminf(fmaxf(sc, 1e-12f), 1.0f);
      float contrib = sl * kInvReg + (-__logf(sc)) * kInvBce;
      if (valid) acc += contrib;
    }
    __syncthreads();
  }

  // ---- wave32 shuffle reduction -> one partial per workgroup ----
#pragma unroll
  for (int off = 16; off > 0; off >>= 1) acc += __shfl_down(acc, off, 32);
  if (lane == 0) partials[blockIdx.x] = acc;
}

// ---------------------------------------------------------------------------
// Kernel 2: 0.3 * sum(pred_scores[:, 8192:]^2) / B   (bandwidth-bound pass)
// ---------------------------------------------------------------------------
__global__ __launch_bounds__(kBlockExtra)
void mask3d_extra_kernel(const float* __restrict__ pred_scores,
                         float* __restrict__ partials) {
  __shared__ float s_red[kBlockExtra];
  const int tid = threadIdx.x;
  const int tail = kNPred - kNGt;
  const int total = kB * tail;
  float acc = 0.0f;
  for (int i = blockIdx.x * kBlockExtra + tid; i < total; i += kGridExtra * kBlockExtra) {
    int b = i / tail;
    int r = i - b * tail;
    float s = pred_scores[(size_t)b * kNPred + kNGt + r];
    acc += s * s;
  }
  s_red[tid] = acc;
  __syncthreads();
  for (int s = kBlockExtra / 2; s > 0; s >>= 1) {
    if (tid < s) s_red[tid] += s_red[tid + s];
    __syncthreads();
  }
  if (tid == 0) partials[kGridMain + blockIdx.x] = s_red[0] * kExtraScale;
}

// ---------------------------------------------------------------------------
// Kernel 3: final deterministic sum of the 4352 partials -> d_out[0]
// ---------------------------------------------------------------------------
__global__ __launch_bounds__(kBlockExtra)
void mask3d_final_kernel(const float* __restrict__ partials,
                         float* __restrict__ out) {
  __shared__ float s_red[kBlockExtra];
  const int tid = threadIdx.x;
  float acc = 0.0f;
  for (int i = tid; i < kNPart; i += kBlockExtra) acc += partials[i];
  s_red[tid] = acc;
  __syncthreads();
  for (int s = kBlockExtra / 2; s > 0; s >>= 1) {
    if (tid < s) s_red[tid] += s_red[tid + s];
    __syncthreads();
  }
  if (tid == 0) out[0] = s_red[0];
}

extern "C" void kernel_launch(void* const* d_in, const int* in_sizes, int n_in,
                              void* d_out, int out_size, void* d_ws, size_t ws_size,
                              hipStream_t stream) {
  const float* pred_boxes  = (const float*)d_in[0];
  const float* pred_scores = (const float*)d_in[1];
  const float* gt_corners  = (const float*)d_in[2];
  float* partials = (float*)d_ws;   // kNPart floats of scratch

  mask3d_gt_kernel<<<kGridMain, kBlock, 0, stream>>>(pred_boxes, pred_scores,
                                                     gt_corners, partials);
  mask3d_extra_kernel<<<kGridExtra, kBlockExtra, 0, stream>>>(pred_scores, partials);
  mask3d_final_kernel<<<1, kBlockExtra, 0, stream>>>(partials, (float*)d_out);
}